// MimicModel_18657337934708
// MI455X (gfx1250) — compile-verified
//
#include <hip/hip_runtime.h>
#include <hip/hip_bf16.h>
#include <stddef.h>

typedef float v2f __attribute__((ext_vector_type(2)));
typedef float v8f __attribute__((ext_vector_type(8)));

#define LD_PAD 64   // padded leading dim for internal node-feature buffers (pow2)

// ---------------------------------------------------------------------------
// Normalization precompute
// ---------------------------------------------------------------------------
__global__ void k_deg_init(float* __restrict__ deg, int n) {
    int i = blockIdx.x * blockDim.x + threadIdx.x;
    if (i < n) deg[i] = 1.0f;   // self-loop weight 1
}

__global__ void k_deg_scatter(const int* __restrict__ col,
                              const float* __restrict__ w,
                              float* __restrict__ deg, int E) {
    int e = blockIdx.x * blockDim.x + threadIdx.x;
    if (e < E) atomicAdd(&deg[col[e]], w[e]);
}

__global__ void k_dinv(const float* __restrict__ deg,
                       float* __restrict__ dinv,
                       float* __restrict__ dinv2, int n) {
    int i = blockIdx.x * blockDim.x + threadIdx.x;
    if (i < n) {
        float d  = deg[i];
        float di = (d > 0.0f) ? rsqrtf(d) : 0.0f;
        dinv[i]  = di;
        dinv2[i] = di * di;
    }
}

__global__ void k_norm(const int* __restrict__ row,
                       const int* __restrict__ col,
                       const float* __restrict__ w,
                       const float* __restrict__ dinv,
                       float* __restrict__ norm, int E) {
    int e = blockIdx.x * blockDim.x + threadIdx.x;
    if (e < E) norm[e] = dinv[row[e]] * w[e] * dinv[col[e]];
}

// ---------------------------------------------------------------------------
// Dense GEMM  H[n, fo] = act(A[n, fi]) @ W[fi, fo]  via V_WMMA_F32_16X16X4_F32
// 256 threads = 8 waves; each wave computes one 16x16 tile; W tile staged in
// LDS zero-padded so the fully-unrolled K loop is branch-free.
// Preconditions: nrows % 16 == 0; A columns [fi, KP) are zero (padded bufs).
// ---------------------------------------------------------------------------
template <int KP, bool RELU>
__global__ __launch_bounds__(256)
void k_gemm_wmma(const float* __restrict__ A, int lda,
                 const float* __restrict__ W, int fi, int fo,
                 float* __restrict__ H, int ldh, int mtiles) {
    __shared__ float Wl[(KP / 2) * 16 * 2];   // [kpair][n16][pair]

    const int tid  = threadIdx.x;
    const int lane = tid & 31;
    const int wave = tid >> 5;
    const int l16  = lane & 15;
    const int half = lane >> 4;               // 0 | 1
    const int ntile = blockIdx.y;

    // ---- stage W tile (bounds-checked once), interleaved for b64 reads ----
    for (int idx = tid; idx < KP * 16; idx += 256) {
        const int k  = idx >> 4;
        const int n  = idx & 15;
        const int nn = ntile * 16 + n;
        float v = 0.0f;
        if (k < fi && nn < fo) v = W[(size_t)k * fo + nn];
        Wl[(((k >> 1) * 16) + n) * 2 + (k & 1)] = v;
    }
    __syncthreads();

    const int mtile = blockIdx.x * 8 + wave;
    if (mtile >= mtiles) return;              // wave-uniform

    const int m = mtile * 16 + l16;
    const float* __restrict__ Arow = A + (size_t)m * lda + half * 2;
    const v2f*   __restrict__ Bv   = ((const v2f*)Wl) + l16;   // [kpair*16]

    v8f acc = {};
    #pragma unroll
    for (int k0 = 0; k0 < KP; k0 += 4) {
        v2f a = *(const v2f*)(Arow + k0);                 // global_load_b64
        if (RELU) { a.x = fmaxf(a.x, 0.0f); a.y = fmaxf(a.y, 0.0f); }
        v2f b = Bv[((k0 >> 1) + half) * 16];              // ds_load_b64
        acc = __builtin_amdgcn_wmma_f32_16x16x4_f32(
            false, a, false, b, (short)0, acc, false, false);
    }

    // D layout: VGPR r -> row (r + half*8), col l16
    const int nn = ntile * 16 + l16;
    if (nn < fo) {
        #pragma unroll
        for (int r = 0; r < 8; ++r) {
            const int mm = mtile * 16 + r + half * 8;
            H[(size_t)mm * ldh + nn] = acc[r];
        }
    }
}

// ---------------------------------------------------------------------------
// out[i,f] = bias[f] + dinv2[i] * H[i,f] for f < fo; 0 for fo <= f < LD_PAD.
// Maintains the zero-padding invariant the next GEMM relies on. ld = 64 (pow2).
// ---------------------------------------------------------------------------
__global__ void k_agg_init(const float* __restrict__ H,
                           const float* __restrict__ dinv2,
                           const float* __restrict__ bias,
                           int n, int fo, float* __restrict__ out) {
    const int t = blockIdx.x * blockDim.x + threadIdx.x;
    const int i = t >> 6;          // / LD_PAD
    const int f = t & (LD_PAD - 1);
    if (i < n) {
        float v = 0.0f;
        if (f < fo) v = bias[f] + dinv2[i] * H[((size_t)i << 6) + f];
        out[((size_t)i << 6) + f] = v;
    }
}

// Final layer (fo == 1) straight into d_out (ld = 1).
__global__ void k_agg_init_last(const float* __restrict__ H,
                                const float* __restrict__ dinv2,
                                const float* __restrict__ bias,
                                int n, float* __restrict__ out) {
    const int i = blockIdx.x * blockDim.x + threadIdx.x;
    if (i < n) out[i] = bias[0] + dinv2[i] * H[(size_t)i << 6];
}

// ---------------------------------------------------------------------------
// Edge scatter: out[col[e], f] += H[row[e], f] * norm[e]
// Consecutive lanes cover consecutive f of the same edge -> coalesced gather
// and contiguous float atomics (resolve in L2 on MI455X).
// ---------------------------------------------------------------------------
__global__ void k_agg_scatter(const float* __restrict__ H, int ldh,
                              const float* __restrict__ norm,
                              const int* __restrict__ row,
                              const int* __restrict__ col,
                              int E, int fo, float* __restrict__ out, int ldo) {
    const long long total  = (long long)E * fo;
    const long long stride = (long long)gridDim.x * blockDim.x;
    for (long long t = blockIdx.x * (long long)blockDim.x + threadIdx.x;
         t < total; t += stride) {
        const int e = (int)(t / fo);
        const int f = (int)(t - (long long)e * fo);
        const float v = H[(size_t)row[e] * ldh + f] * norm[e];
        atomicAdd(&out[(size_t)col[e] * ldo + f], v);
    }
}

// ---------------------------------------------------------------------------
static void launch_gemm(int layer, const float* A, int lda, const float* W,
                        int fi, int fo, float* H, int mtiles, hipStream_t s) {
    dim3 grid((mtiles + 7) / 8, (fo + 15) / 16);
    if (layer == 0) {
        k_gemm_wmma<128, false><<<grid, 256, 0, s>>>(A, lda, W, fi, fo, H, LD_PAD, mtiles);
    } else if (fi == 50) {
        k_gemm_wmma<52, true><<<grid, 256, 0, s>>>(A, lda, W, fi, fo, H, LD_PAD, mtiles);
    } else if (fi == 30) {
        k_gemm_wmma<32, true><<<grid, 256, 0, s>>>(A, lda, W, fi, fo, H, LD_PAD, mtiles);
    } else { // fi == 10
        k_gemm_wmma<12, true><<<grid, 256, 0, s>>>(A, lda, W, fi, fo, H, LD_PAD, mtiles);
    }
}

extern "C" void kernel_launch(void* const* d_in, const int* in_sizes, int n_in,
                              void* d_out, int out_size, void* d_ws, size_t ws_size,
                              hipStream_t stream) {
    (void)n_in; (void)ws_size; (void)out_size;

    const float* x  = (const float*)d_in[0];
    const int*   ei = (const int*)d_in[1];
    const float* ew = (const float*)d_in[2];
    const float* Wp[7]; const float* Bp[7];
    for (int i = 0; i < 7; ++i) {
        Wp[i] = (const float*)d_in[3 + 2 * i];
        Bp[i] = (const float*)d_in[4 + 2 * i];
    }

    const int NF = 128;
    const int N  = in_sizes[0] / NF;       // 100000 (multiple of 16)
    const int E  = in_sizes[2];            // 1600000
    const int* row = ei;                   // edge_index[0]
    const int* col = ei + E;               // edge_index[1]

    // workspace carve-up (floats)
    float* ws    = (float*)d_ws;
    float* deg   = ws;
    float* dinv  = deg   + N;
    float* dinv2 = dinv  + N;
    float* norm  = dinv2 + N;
    float* hbuf  = norm  + E;              // N x LD_PAD
    float* abuf  = hbuf  + (size_t)N * LD_PAD;

    const int T = 256;
    // ---- shared normalization (computed once, used by all 7 layers) ----
    k_deg_init   <<<(N + T - 1) / T, T, 0, stream>>>(deg, N);
    k_deg_scatter<<<(E + T - 1) / T, T, 0, stream>>>(col, ew, deg, E);
    k_dinv       <<<(N + T - 1) / T, T, 0, stream>>>(deg, dinv, dinv2, N);
    k_norm       <<<(E + T - 1) / T, T, 0, stream>>>(row, col, ew, dinv, norm, E);

    const int fis[7] = {128, 50, 50, 30, 30, 10, 10};
    const int fos[7] = { 50, 50, 30, 30, 10, 10,  1};

    const float* cur = x;
    int ldcur = NF;
    const int mtiles = N / 16;             // exact

    for (int l = 0; l < 7; ++l) {
        const int fi = fis[l], fo = fos[l];

        launch_gemm(l, cur, ldcur, Wp[l], fi, fo, hbuf, mtiles, stream);

        if (l == 6) {
            k_agg_init_last<<<(N + T - 1) / T, T, 0, stream>>>(hbuf, dinv2, Bp[l],
                                                               N, (float*)d_out);
            k_agg_scatter  <<<4096, T, 0, stream>>>(hbuf, LD_PAD, norm, row, col,
                                                    E, 1, (float*)d_out, 1);
        } else {
            k_agg_init <<<((size_t)N * LD_PAD + T - 1) / T, T, 0, stream>>>(
                hbuf, dinv2, Bp[l], N, fo, abuf);
            k_agg_scatter<<<4096, T, 0, stream>>>(hbuf, LD_PAD, norm, row, col,
                                                  E, fo, abuf, LD_PAD);
        }

        cur = abuf; ldcur = LD_PAD;
    }
}